// SplitNN_78623671320711
// MI455X (gfx1250) — compile-verified
//
#include <hip/hip_runtime.h>

typedef __attribute__((ext_vector_type(2))) float v2f;
typedef __attribute__((ext_vector_type(4))) float v4f;
typedef __attribute__((ext_vector_type(8))) float v8f;

#define B_ROWS 262144
#define F_DIM  160
#define H_DIM  128
#define C_DIM  10
#define KP_F   (F_DIM / 2)   // 80 k-pairs in the feature dim
#define KP_H   (H_DIM / 2)   // 64 k-pairs in the hidden dim
#define INV_EPS 7.5f         // 1 / (4/30)
#define P_KEEP  0.95f

__device__ __forceinline__ v8f wmma_f32(v2f a, v2f b, v8f c) {
  // D = A(16x4,f32) * B(4x16,f32) + C(16x16,f32)
  return __builtin_amdgcn_wmma_f32_16x16x4_f32(
      /*neg_a=*/false, a, /*neg_b=*/false, b,
      /*c_mod=*/(short)0, c, /*reuse_a=*/false, /*reuse_b=*/false);
}

__device__ __forceinline__ v8f splat8(float v) {
  v8f r;
#pragma unroll
  for (int i = 0; i < 8; ++i) r[i] = v;
  return r;
}

// ---------------------------------------------------------------------------
// Pass 0: init sensitivity max/min slots.
// ---------------------------------------------------------------------------
__global__ void k_init(int* __restrict__ stat) {
  stat[0] = 0;            // running max of L1 norms (norms >= 0)
  stat[1] = 0x7F800000;   // +inf as running min
}

// ---------------------------------------------------------------------------
// Pass 1: per-row L1 norm of out1 = x1@W1 + b1 -> global max/min atomics.
// W1 staged in LDS pair-packed so B-frags are single ds_load_b64.
// A-frags software-pipelined one k-step ahead.
// ---------------------------------------------------------------------------
__global__ void __launch_bounds__(256) k_norm(const float* __restrict__ x1,
                                              const float* __restrict__ W1,
                                              const float* __restrict__ b1,
                                              int* __restrict__ stat) {
  __shared__ v2f w1p[KP_F * H_DIM];   // 80 KB: {W1[2kp][h], W1[2kp+1][h]}
  __shared__ float smax[8];
  __shared__ float smin[8];

  for (int idx = threadIdx.x; idx < KP_F * H_DIM; idx += 256) {
    const int kp = idx >> 7, h = idx & (H_DIM - 1);
    v2f w;
    w.x = W1[(2 * kp) * H_DIM + h];
    w.y = W1[(2 * kp + 1) * H_DIM + h];
    w1p[idx] = w;
  }
  __syncthreads();

  const int lane = threadIdx.x & 31;
  const int wave = threadIdx.x >> 5;          // wave32
  const int n    = lane & 15;                 // N (C/D layout), also A-row M
  const int kg   = lane >> 4;                 // K-pair selector (A/B layout)
  const int row0 = (blockIdx.x * 8 + wave) * 16;
  const int wb   = kg * H_DIM + n;            // loop-invariant per-lane part of B index

  v8f acc[8];
#pragma unroll
  for (int t = 0; t < 8; ++t) acc[t] = splat8(b1[t * 16 + n]);

  const v2f* av = (const v2f*)(x1 + (size_t)(row0 + n) * F_DIM);
  v2f a = av[kg];                                    // k = 0 fragment
  for (int k = 0; k < F_DIM; k += 4) {
    const int kn = (k + 4 < F_DIM) ? (k + 4) : 0;    // prefetch next A frag
    const v2f anext = av[(kn >> 1) + kg];
    const v2f* wrow = w1p + (k >> 1) * H_DIM + wb;
#pragma unroll
    for (int t = 0; t < 8; ++t) acc[t] = wmma_f32(a, wrow[t * 16], acc[t]);
    a = anext;
  }

  // Per-lane partial |.| sums; VGPR r holds row (r + 8*kg), column n.
  float s[8];
#pragma unroll
  for (int r = 0; r < 8; ++r) {
    float v = 0.0f;
#pragma unroll
    for (int t = 0; t < 8; ++t) v += fabsf(acc[t][r]);
    s[r] = v;
  }
#pragma unroll
  for (int off = 1; off <= 8; off <<= 1) {
#pragma unroll
    for (int r = 0; r < 8; ++r) s[r] += __shfl_xor(s[r], off, 32);
  }
  float mx = s[0], mn = s[0];
#pragma unroll
  for (int r = 1; r < 8; ++r) { mx = fmaxf(mx, s[r]); mn = fminf(mn, s[r]); }
  mx = fmaxf(mx, __shfl_xor(mx, 16, 32));
  mn = fminf(mn, __shfl_xor(mn, 16, 32));

  if (lane == 0) { smax[wave] = mx; smin[wave] = mn; }
  __syncthreads();
  if (threadIdx.x == 0) {
    float bmx = smax[0], bmn = smin[0];
    for (int w = 1; w < 8; ++w) {
      bmx = fmaxf(bmx, smax[w]);
      bmn = fminf(bmn, smin[w]);
    }
    atomicMax(&stat[0], __float_as_int(bmx));   // non-negative: int-ordered
    atomicMin(&stat[1], __float_as_int(bmn));
  }
}

// ---------------------------------------------------------------------------
// Pass 2: fused out1/out2 GEMMs + noise + RR mask + min + 128x10 head GEMM.
// W1/W2 interleaved pair-packed in LDS: one ds_load_b128 feeds both WMMAs.
// ---------------------------------------------------------------------------
__global__ void __launch_bounds__(256) k_main(
    const float* __restrict__ x1, const float* __restrict__ x2,
    const float* __restrict__ W1, const float* __restrict__ b1,
    const float* __restrict__ W2, const float* __restrict__ b2,
    const float* __restrict__ Ws, const float* __restrict__ bs,
    const float* __restrict__ noise, const float* __restrict__ rr,
    const int* __restrict__ stat, float* __restrict__ out) {
  __shared__ v4f  wq[KP_F * H_DIM];    // 160 KB: {W1[2kp][h],W1[2kp+1][h],W2[..],W2[..]}
  __shared__ v2f  wspp[KP_H * 16];     // 8 KB: Ws pair-packed, zero-padded to 16 cols
  __shared__ float scr[8][16 * H_DIM]; // 64 KB: per-wave server_input tile

  const int lane = threadIdx.x & 31;
  const int wave = threadIdx.x >> 5;
  const int n    = lane & 15;
  const int kg   = lane >> 4;
  const int row0 = (blockIdx.x * 8 + wave) * 16;

  // Pull the epilogue-only noise tile toward L2 while the GEMM runs.
  __builtin_prefetch(noise + (size_t)row0 * H_DIM + lane * 64, 0, 0);

  for (int idx = threadIdx.x; idx < KP_F * H_DIM; idx += 256) {
    const int kp = idx >> 7, h = idx & (H_DIM - 1);
    v4f w;
    w.x = W1[(2 * kp) * H_DIM + h];
    w.y = W1[(2 * kp + 1) * H_DIM + h];
    w.z = W2[(2 * kp) * H_DIM + h];
    w.w = W2[(2 * kp + 1) * H_DIM + h];
    wq[idx] = w;
  }
  for (int idx = threadIdx.x; idx < KP_H * 16; idx += 256) {
    const int kp = idx >> 4, c = idx & 15;
    v2f w;
    w.x = (c < C_DIM) ? Ws[(2 * kp) * C_DIM + c] : 0.0f;
    w.y = (c < C_DIM) ? Ws[(2 * kp + 1) * C_DIM + c] : 0.0f;
    wspp[idx] = w;
  }
  __syncthreads();

  const float noise_mul =
      (__int_as_float(stat[0]) - __int_as_float(stat[1])) * INV_EPS;

  v8f acc1[8], acc2[8];
#pragma unroll
  for (int t = 0; t < 8; ++t) {
    acc1[t] = splat8(b1[t * 16 + n]);
    acc2[t] = splat8(b2[t * 16 + n]);
  }

  const v2f* a1v = (const v2f*)(x1 + (size_t)(row0 + n) * F_DIM);
  const v2f* a2v = (const v2f*)(x2 + (size_t)(row0 + n) * F_DIM);
  const int  wb  = kg * H_DIM + n;

  v2f a1 = a1v[kg], a2 = a2v[kg];
  for (int k = 0; k < F_DIM; k += 4) {
    const int kn = (k + 4 < F_DIM) ? (k + 4) : 0;
    const v2f na1 = a1v[(kn >> 1) + kg];
    const v2f na2 = a2v[(kn >> 1) + kg];
    const v4f* wrow = wq + (k >> 1) * H_DIM + wb;
#pragma unroll
    for (int t = 0; t < 8; ++t) {
      const v4f w = wrow[t * 16];
      v2f bw1; bw1.x = w.x; bw1.y = w.y;
      v2f bw2; bw2.x = w.z; bw2.y = w.w;
      acc1[t] = wmma_f32(a1, bw1, acc1[t]);
      acc2[t] = wmma_f32(a2, bw2, acc2[t]);
    }
    a1 = na1; a2 = na2;
  }

  // Epilogue: DP noise, random-response mask, elementwise min; stash in LDS
  // row-major [m][h] so the head GEMM can reload it in WMMA A-layout.
  float* sc = scr[wave];
#pragma unroll
  for (int t = 0; t < 8; ++t) {
    const float maskv = (rr[t * 16 + n] < P_KEEP) ? 1.0f : 0.0f;
#pragma unroll
    for (int r = 0; r < 8; ++r) {
      const int mrow = r + 8 * kg;
      const float nz = noise[(size_t)(row0 + mrow) * H_DIM + t * 16 + n];
      const float o1 = (acc1[t][r] + nz * noise_mul) * maskv;
      sc[mrow * H_DIM + t * 16 + n] = fminf(o1, acc2[t][r]);
    }
  }

  // Head GEMM: [16x128] @ [128x16-padded] via 32 fp32 WMMAs, all from LDS.
  v8f accS = splat8((n < C_DIM) ? bs[n] : 0.0f);
  const v2f* scv = (const v2f*)sc;
  for (int k = 0; k < H_DIM; k += 4) {
    const v2f a = scv[n * (H_DIM / 2) + (k >> 1) + kg];
    const v2f b = wspp[((k >> 1) + kg) * 16 + n];
    accS = wmma_f32(a, b, accS);
  }

  if (n < C_DIM) {
#pragma unroll
    for (int r = 0; r < 8; ++r) {
      const int mrow = r + 8 * kg;
      out[(size_t)(row0 + mrow) * C_DIM + n] = accS[r];
    }
  }
}

// ---------------------------------------------------------------------------
extern "C" void kernel_launch(void* const* d_in, const int* in_sizes, int n_in,
                              void* d_out, int out_size, void* d_ws,
                              size_t ws_size, hipStream_t stream) {
  const float* x1    = (const float*)d_in[0];
  const float* x2    = (const float*)d_in[1];
  const float* W1    = (const float*)d_in[2];
  const float* b1    = (const float*)d_in[3];
  const float* W2    = (const float*)d_in[4];
  const float* b2    = (const float*)d_in[5];
  const float* Ws    = (const float*)d_in[6];
  const float* bs    = (const float*)d_in[7];
  const float* noise = (const float*)d_in[8];
  const float* rr    = (const float*)d_in[9];
  float* out = (float*)d_out;
  int* stat  = (int*)d_ws;

  k_init<<<1, 1, 0, stream>>>(stat);
  // 8 waves/block, 16 rows/wave -> 128 rows/block
  k_norm<<<B_ROWS / 128, 256, 0, stream>>>(x1, W1, b1, stat);
  k_main<<<B_ROWS / 128, 256, 0, stream>>>(x1, x2, W1, b1, W2, b2, Ws, bs,
                                           noise, rr, stat, out);
}